// DrawInstance_36807869727196
// MI455X (gfx1250) — compile-verified
//
#include <hip/hip_runtime.h>

typedef __attribute__((ext_vector_type(2))) float v2f;
typedef __attribute__((ext_vector_type(8))) float v8f;

#define Bn    8
#define Nn    100
#define HWn   65536   // 256*256
#define Cn    16
#define ALPHAv 0.3f

// One wave computes agg[16 pixels][16 classes] = masks_tile(16xK) x onehot(Kx16)
// with V_WMMA_F32_16X16X4_F32, K looped 25x over the 100 detections.
__global__ __launch_bounds__(256)
void draw_instance_wmma(const float* __restrict__ images,
                        const int*   __restrict__ det_outs,
                        const float* __restrict__ masks,
                        const float* __restrict__ colors,
                        float*       __restrict__ out)
{
    __shared__ int   s_cls[Nn];       // class id per detection for this batch
    __shared__ float s_col[Cn * 3];   // colors pre-scaled by ALPHA
    __shared__ float s_agg[8 * 256];  // per-wave 16x16 f32 tile

    const int tid  = threadIdx.x;
    const int wave = tid >> 5;
    const int lane = tid & 31;

    const int b      = (int)(blockIdx.x >> 9);   // 512 blocks per batch image
    const int blkInB = (int)(blockIdx.x & 511);

    if (tid < Nn)      s_cls[tid] = det_outs[(b * Nn + tid) * 6 + 4]; // index -2 of 6
    if (tid < Cn * 3)  s_col[tid] = colors[tid] * ALPHAv;
    __syncthreads();

    const int tile     = blkInB * 8 + wave;      // 0..4095 within batch
    const int pix_base = tile * 16;
    const int m        = lane & 15;              // A-layout: lane -> M (pixel row)
    const int kh       = (lane >> 4) << 1;       // upper half-wave holds K+2,K+3

    // This lane's pixel column within the mask planes of batch b.
    const float* mp = masks + (size_t)b * Nn * HWn + (size_t)(pix_base + m);

    v8f acc0 = {};
    v8f acc1 = {};
#pragma unroll
    for (int i = 0; i < 25; ++i) {               // 25 * K4 = 100 detections
        const int ka = i * 4 + kh;
        // A: 16x4 f32 tile (mask values); two planes per lane, 64B coalesced per half-wave
        v2f a;
        a.x = mp[(size_t)ka * HWn];
        a.y = mp[(size_t)(ka + 1) * HWn];
        // B: 4x16 one-hot; lane = class column, rows selected by kh like A's K split
        v2f bb;
        bb.x = (s_cls[ka]     == m) ? 1.0f : 0.0f;
        bb.y = (s_cls[ka + 1] == m) ? 1.0f : 0.0f;
        if (i & 1)
            acc1 = __builtin_amdgcn_wmma_f32_16x16x4_f32(false, a, false, bb,
                                                         (short)0, acc1, false, false);
        else
            acc0 = __builtin_amdgcn_wmma_f32_16x16x4_f32(false, a, false, bb,
                                                         (short)0, acc0, false, false);
    }

    // C/D layout: VGPR v -> M = v + 8*(lane>=16), lane&15 -> class column
#pragma unroll
    for (int v = 0; v < 8; ++v) {
        const int mrow = v + ((lane >> 4) << 3);
        s_agg[wave * 256 + mrow * 16 + (lane & 15)] = acc0[v] + acc1[v];
    }
    __syncthreads();

    if (lane < 16) {
        const int pix = pix_base + lane;
        const float* arow = &s_agg[wave * 256 + lane * 16];
        float r = 0.0f, g = 0.0f, bl = 0.0f;
#pragma unroll
        for (int c = 0; c < Cn; ++c) {
            const float s = (arow[c] > 0.5f) ? 1.0f : 0.0f;  // seg threshold
            r  = fmaf(s, s_col[c * 3 + 0], r);
            g  = fmaf(s, s_col[c * 3 + 1], g);
            bl = fmaf(s, s_col[c * 3 + 2], bl);
        }
        const size_t ib = ((size_t)b * HWn + (size_t)pix) * 3;
        float o0 = fminf(fmaxf(images[ib + 0] + r,  0.0f), 255.0f);
        float o1 = fminf(fmaxf(images[ib + 1] + g,  0.0f), 255.0f);
        float o2 = fminf(fmaxf(images[ib + 2] + bl, 0.0f), 255.0f);
        out[ib + 0] = floorf(o0);   // uint8 cast truncates (values >= 0)
        out[ib + 1] = floorf(o1);
        out[ib + 2] = floorf(o2);
    }
}

extern "C" void kernel_launch(void* const* d_in, const int* in_sizes, int n_in,
                              void* d_out, int out_size, void* d_ws, size_t ws_size,
                              hipStream_t stream) {
    (void)in_sizes; (void)n_in; (void)d_ws; (void)ws_size; (void)out_size;
    const float* images = (const float*)d_in[0];
    const int*   det    = (const int*)  d_in[1];
    const float* masks  = (const float*)d_in[2];
    const float* colors = (const float*)d_in[3];
    float* out = (float*)d_out;

    // 8 batches * 512 blocks, 8 waves/block, one 16-pixel tile per wave
    dim3 grid(Bn * 512);
    dim3 block(256);
    draw_instance_wmma<<<grid, block, 0, stream>>>(images, det, masks, colors, out);
}